// QuantumAttentionMechanism_90280212562551
// MI455X (gfx1250) — compile-verified
//
#include <hip/hip_runtime.h>
#include <math.h>

typedef __attribute__((ext_vector_type(2))) float v2f;
typedef __attribute__((ext_vector_type(8))) float v8f;

#define NQ 4
#define INV_SQRT2 0.70710678118654752440f
#define HALF_PI   1.57079632679489661923f

// ws layout (floats):
//   [0]            softmax denominator accumulator
//   [8..39]        Rot gates: per wire w, 8 floats {g00r,g00i,g01r,g01i,g10r,g10i,g11r,g11i}
//   [64..64+D)     zero row (B operand for WMMA pad columns 8-15)
//   [320 + n*16]   per-node table, 16 floats/node:
//        [0..7]  query wires: (h0,h1) per wire  (post RY+H separable factors)
//        [8..15] key wires:   (cos(t/2), sin(t/2)) per wire
#define GATES_OFF    8
#define ZERO_OFF     64
#define NODE_TAB_OFF 320

__global__ void qa_init(float* __restrict__ ws, const float* __restrict__ qp, int D) {
    const int tid = threadIdx.x;
    if (tid == 0) ws[0] = 0.0f;
    for (int i = tid; i < D; i += blockDim.x) ws[ZERO_OFF + i] = 0.0f;
    if (tid < NQ) {
        const int w = tid;
        const float phi = qp[3*w + 0], th = qp[3*w + 1], om = qp[3*w + 2];
        float sep, cep, sem, cem, st, ct;
        sincosf(0.5f * (phi + om), &sep, &cep);
        sincosf(0.5f * (phi - om), &sem, &cem);
        sincosf(0.5f * th,          &st,  &ct);
        float* g = ws + GATES_OFF + 8 * w;
        g[0] =  cep * ct; g[1] = -sep * ct;   // g00
        g[2] = -cem * st; g[3] = -sem * st;   // g01
        g[4] =  cem * st; g[5] = -sem * st;   // g10
        g[6] =  cep * ct; g[7] =  sep * ct;   // g11
    }
}

// One wave = 16 nodes. C tile = 16 nodes x 16 cols; cols 0-3 = q, 4-7 = k, 8-15 pad (zero B).
// A (16x4 f32): lanes 0-15 hold M=lane, VGPR0=K0,VGPR1=K1; lanes 16-31: K2,K3.
// B (4x16 f32): lanes 0-15 hold N=lane, VGPR0=K0,VGPR1=K1; lanes 16-31: K2,K3.
__global__ void qa_node_kernel(const float* __restrict__ X,
                               const float* __restrict__ Wq,
                               const float* __restrict__ bq,
                               const float* __restrict__ Wk,
                               const float* __restrict__ bk,
                               const float* __restrict__ zeroRow,
                               float* __restrict__ nodeTab,
                               int N, int D) {
    const int lane = threadIdx.x & 31;
    const int wid  = threadIdx.x >> 5;
    const int tileBase = (blockIdx.x * (blockDim.x >> 5) + wid) * 16;
    if (tileBase >= N) return;              // wave-uniform: EXEC full below

    const int n     = lane & 15;            // B column / A row-within-tile
    const int khalf = (lane >> 4) * 2;      // 0 or 2

    int arow = tileBase + n;
    if (arow >= N) arow = N - 1;            // clamp (duplicate load, stores masked)
    const float* aptr = X + (size_t)arow * D + khalf;

    // pad lanes read a real zero row: uniform control flow, no EXEC juggling
    const float* bptr = (n < 4) ? (Wq + n * D)
                     : (n < 8) ? (Wk + (n - 4) * D)
                               : zeroRow;
    bptr += khalf;

    v8f acc = {0.f,0.f,0.f,0.f,0.f,0.f,0.f,0.f};
    #pragma unroll 8
    for (int kk = 0; kk < D; kk += 4) {
        const v2f a = *(const v2f*)(aptr + kk);
        const v2f b = *(const v2f*)(bptr + kk);
        acc = __builtin_amdgcn_wmma_f32_16x16x4_f32(
                  false, a, false, b, (short)0, acc, false, false);
    }

    // C layout: lane l holds N=l&15, M = r + 8*(l>=16) for VGPR r
    const int moff = (lane >> 4) * 8;
    #pragma unroll
    for (int r = 0; r < 8; ++r) {
        const int node = tileBase + r + moff;
        if (node >= N) continue;
        const float v = acc[r];
        if (n < 4) {                         // query wire n
            const float t = tanhf(v + bq[n]) * HALF_PI;
            float s, c; sincosf(0.5f * t, &s, &c);
            float2 h = make_float2((c + s) * INV_SQRT2, (c - s) * INV_SQRT2);
            *(float2*)(nodeTab + (size_t)node * 16 + 2 * n) = h;
        } else if (n < 8) {                  // key wire n-4
            const int w = n - 4;
            const float t = tanhf(v + bk[w]) * HALF_PI;
            float s, c; sincosf(0.5f * t, &s, &c);
            *(float2*)(nodeTab + (size_t)node * 16 + 8 + 2 * w) = make_float2(c, s);
        }
    }
}

// wire w <-> bit (3-w): flat index b = b0*8 + b1*4 + b2*2 + b3
__global__ void qa_edge_kernel(const int* __restrict__ eidx,
                               const float* __restrict__ nodeTab,
                               const float* __restrict__ gates,  // ws+GATES_OFF, uniform
                               float* __restrict__ out,
                               float* __restrict__ gsum,
                               int E) {
    const int e = blockIdx.x * blockDim.x + threadIdx.x;
    float z = 0.0f;
    if (e < E) {
        const int src = eidx[e];
        const int dst = eidx[E + e];
        const float4 q0 = *(const float4*)(nodeTab + (size_t)src * 16);
        const float4 q1 = *(const float4*)(nodeTab + (size_t)src * 16 + 4);
        const float4 k0 = *(const float4*)(nodeTab + (size_t)dst * 16 + 8);
        const float4 k1 = *(const float4*)(nodeTab + (size_t)dst * 16 + 12);
        const float h[4][2] = {{q0.x,q0.y},{q0.z,q0.w},{q1.x,q1.y},{q1.z,q1.w}};
        const float co[4] = {k0.x, k0.z, k1.x, k1.z};
        const float si[4] = {k0.y, k0.w, k1.y, k1.w};

        // real separable state after RY + H layers
        float s[16];
        #pragma unroll
        for (int b = 0; b < 16; ++b)
            s[b] = h[0][(b>>3)&1] * h[1][(b>>2)&1] * h[2][(b>>1)&1] * h[3][b&1];

        // CRY(k_w) on (control w, target (w+1)%4): still real
        #pragma unroll
        for (int w = 0; w < 4; ++w) {
            const int cb = 1 << (3 - w);
            const int tb = 1 << (3 - ((w + 1) & 3));
            #pragma unroll
            for (int m = 0; m < 16; ++m) {
                if ((m & cb) && !(m & tb)) {
                    const float a = s[m], b_ = s[m | tb];
                    s[m]      = co[w] * a - si[w] * b_;
                    s[m | tb] = si[w] * a + co[w] * b_;
                }
            }
        }

        // Rot(phi,theta,omega) per wire: gates precomputed (uniform -> scalar loads)
        float sr[16], sim[16];
        #pragma unroll
        for (int b = 0; b < 16; ++b) { sr[b] = s[b]; sim[b] = 0.f; }

        #pragma unroll
        for (int w = 0; w < 4; ++w) {
            const float* g = gates + 8 * w;
            const float g00r = g[0], g00i = g[1];
            const float g01r = g[2], g01i = g[3];
            const float g10r = g[4], g10i = g[5];
            const float g11r = g[6], g11i = g[7];
            const int tb = 1 << (3 - w);
            #pragma unroll
            for (int m = 0; m < 16; ++m) {
                if (!(m & tb)) {
                    const float ar = sr[m],      ai = sim[m];
                    const float br = sr[m | tb], bi = sim[m | tb];
                    sr[m]       = g00r*ar - g00i*ai + g01r*br - g01i*bi;
                    sim[m]      = g00r*ai + g00i*ar + g01r*bi + g01i*br;
                    sr[m | tb]  = g10r*ar - g10i*ai + g11r*br - g11i*bi;
                    sim[m | tb] = g10r*ai + g10i*ar + g11r*bi + g11i*br;
                }
            }
        }

        // mean_w <Z_w> = (1/4) * sum_b p[b] * (4 - 2*popcount(b))
        float attn = 0.f;
        #pragma unroll
        for (int b = 0; b < 16; ++b) {
            const int pc = ((b>>3)&1) + ((b>>2)&1) + ((b>>1)&1) + (b&1);
            const float p = sr[b]*sr[b] + sim[b]*sim[b];
            attn += p * (1.0f - 0.5f * (float)pc);
        }
        z = expf(attn);               // attn in [-1,1]: no max-shift needed
        out[e] = z;
    }

    // block reduction of softmax denominator
    __shared__ float blkSum;
    if (threadIdx.x == 0) blkSum = 0.f;
    __syncthreads();
    float t = z;
    #pragma unroll
    for (int off = 16; off > 0; off >>= 1) t += __shfl_down(t, off, 32);
    if ((threadIdx.x & 31) == 0) atomicAdd(&blkSum, t);
    __syncthreads();
    if (threadIdx.x == 0) atomicAdd(gsum, blkSum);
}

__global__ void qa_scale(float* __restrict__ out, const float* __restrict__ gsum, int E) {
    const int e = blockIdx.x * blockDim.x + threadIdx.x;
    const float r = 1.0f / gsum[0];   // uniform -> scalar load
    if (e < E) out[e] *= r;
}

extern "C" void kernel_launch(void* const* d_in, const int* in_sizes, int n_in,
                              void* d_out, int out_size, void* d_ws, size_t ws_size,
                              hipStream_t stream) {
    const float* X  = (const float*)d_in[0];
    const int*   ei = (const int*)  d_in[1];
    const float* Wq = (const float*)d_in[2];
    const float* bq = (const float*)d_in[3];
    const float* Wk = (const float*)d_in[4];
    const float* bk = (const float*)d_in[5];
    const float* qp = (const float*)d_in[6];

    const int D = in_sizes[2] / NQ;      // 256
    const int N = in_sizes[0] / D;       // 50000
    const int E = in_sizes[1] / 2;       // 800000

    float* ws      = (float*)d_ws;
    float* nodeTab = ws + NODE_TAB_OFF;
    float* out     = (float*)d_out;

    qa_init<<<1, 256, 0, stream>>>(ws, qp, D);

    const int nodesPerBlock = 128;       // 8 waves x 16 nodes
    qa_node_kernel<<<(N + nodesPerBlock - 1) / nodesPerBlock, 256, 0, stream>>>(
        X, Wq, bq, Wk, bk, ws + ZERO_OFF, nodeTab, N, D);

    const int eb = (E + 255) / 256;
    qa_edge_kernel<<<eb, 256, 0, stream>>>(ei, nodeTab, ws + GATES_OFF, out, ws, E);
    qa_scale<<<eb, 256, 0, stream>>>(out, ws, E);
}